// L1OutUB_26242250178718
// MI455X (gfx1250) — compile-verified
//
#include <hip/hip_runtime.h>
#include <math.h>

typedef __attribute__((ext_vector_type(2))) float v2f;
typedef __attribute__((ext_vector_type(4))) float v4f;
typedef __attribute__((ext_vector_type(8))) float v8f;
typedef __attribute__((ext_vector_type(4))) int   v4i;

#define BATCH 512
#define XD 768
#define YD 128

// workspace layout (float elements):
//   IV  [512*128] @ 0        : exp(-logvar)
//   M2  [512*128] @ 65536    : -2 * mu * exp(-logvar)
//   acc [4]       @ 131072   : [0]=sum_{i,j} pair GEMM, [1]=sum_i(C+L), [2]=sum_i positive
#define IV_OFF 0
#define M2_OFF (512 * 128)
#define ACC_OFF (2 * 512 * 128)

static __device__ __forceinline__ v8f wmma_f32(v2f a, v2f b, v8f c) {
    // D = A(16x4,f32) * B(4x16,f32) + C(16x16,f32); fp32-exact WMMA on gfx1250
    return __builtin_amdgcn_wmma_f32_16x16x4_f32(false, a, false, b, (short)0, c,
                                                 false, false);
}

// ---- CDNA5 async global->LDS staging (ASYNCcnt path), with portable fallback ----
#if __has_builtin(__builtin_amdgcn_global_load_async_to_lds_b128)
#define HAVE_ASYNC_LDS 1
static __device__ __forceinline__ void async_tile_16x128(const float* __restrict__ g,
                                                         float* __restrict__ s, int lane) {
    // one contiguous 8 KB tile: 512 x B128 lane-transfers, 16 rounds of 32 lanes
    for (int c = lane; c < 512; c += 32) {
        __builtin_amdgcn_global_load_async_to_lds_b128((v4i*)(g + c * 4),
                                                       (v4i*)(s + c * 4), 0, 0);
    }
}
static __device__ __forceinline__ void async_wait0() {
#if __has_builtin(__builtin_amdgcn_s_wait_asynccnt)
    __builtin_amdgcn_s_wait_asynccnt(0);
#else
    asm volatile("s_wait_asynccnt 0x0" ::: "memory");
#endif
}
#else
#define HAVE_ASYNC_LDS 0
static __device__ __forceinline__ void async_tile_16x128(const float* __restrict__ g,
                                                         float* __restrict__ s, int lane) {
    for (int c = lane; c < 512; c += 32)
        *(v4f*)(s + c * 4) = *(const v4f*)(g + c * 4);
}
static __device__ __forceinline__ void async_wait0() {}
#endif

// ---------------------------------------------------------------------------
// Kernel 1: fused dual-MLP for 16 samples per block (one wave).
//   layer1: x[16,768] @ [w1_mu | w1_lv][768,16]  -> 192 WMMAs
//   layer2: h_mu[16,8] @ w2_mu[8,128], h_lv[16,8] @ w2_lv[8,128] -> 32 WMMAs
// Emits IV, M2 rows + scalar partial sums (positive, C+L).
// ---------------------------------------------------------------------------
__global__ __launch_bounds__(32) void mlp_kernel(
    const float* __restrict__ x, const float* __restrict__ y,
    const float* __restrict__ w1mu, const float* __restrict__ b1mu,
    const float* __restrict__ w2mu, const float* __restrict__ b2mu,
    const float* __restrict__ w1lv, const float* __restrict__ b1lv,
    const float* __restrict__ w2lv, const float* __restrict__ b2lv,
    float* __restrict__ IV, float* __restrict__ M2, float* __restrict__ acc) {
    __shared__ float hs[16 * 16];    // relu(x@W1 + b1), cols 0-7 = mu-branch, 8-15 = lv-branch
    __shared__ float mus[16 * YD];   // mu tile [16,128]
    __shared__ float lvs[16 * YD];   // tanh(logvar) tile [16,128]

    const int l  = threadIdx.x;   // 0..31 (wave32)
    const int m  = l & 15;        // row (A) or col (B/C) within 16x16 fragment
    const int hi = l >> 4;        // half-wave select for K pairs
    const int r0 = blockIdx.x * 16;

    const float* xrow = x + (size_t)(r0 + m) * XD;
    __builtin_prefetch(xrow, 0, 3);  // -> global_prefetch_b8

    // ---- layer 1: accumulate over K=768 in steps of 4 ----
    v8f c1 = {};
    const float* wsel = (m < 8) ? w1mu : w1lv;  // lane-local select, no EXEC divergence
    const int wcol = m & 7;
    for (int k0 = 0; k0 < XD; k0 += 4) {
        const int kk = k0 + 2 * hi;
        v2f a;
        a.x = xrow[kk];
        a.y = xrow[kk + 1];
        v2f b;
        b.x = wsel[(size_t)kk * 8 + wcol];
        b.y = wsel[(size_t)(kk + 1) * 8 + wcol];
        c1 = wmma_f32(a, b, c1);
    }
    const float bias1 = (m < 8) ? b1mu[m] : b1lv[m - 8];
    for (int v = 0; v < 8; ++v) {
        const int mm = v + 8 * hi;                       // C layout: M = vgpr + 8*(lane>=16)
        hs[mm * 16 + m] = fmaxf(c1[v] + bias1, 0.0f);    // ReLU
    }
    __syncthreads();

    // ---- layer 2: both heads, 8 N-tiles of 16 cols each ----
    for (int nt = 0; nt < 8; ++nt) {
        v8f cm = {};
        v8f cl = {};
        const int col = nt * 16 + m;  // output column for B/C fragments
        for (int k0 = 0; k0 < 8; k0 += 4) {
            const int kk = k0 + 2 * hi;
            v2f am, al, bm, bl;
            am.x = hs[m * 16 + kk];       // h_mu = cols 0..7
            am.y = hs[m * 16 + kk + 1];
            al.x = hs[m * 16 + 8 + kk];   // h_lv = cols 8..15
            al.y = hs[m * 16 + 8 + kk + 1];
            bm.x = w2mu[(size_t)kk * YD + col];
            bm.y = w2mu[(size_t)(kk + 1) * YD + col];
            bl.x = w2lv[(size_t)kk * YD + col];
            bl.y = w2lv[(size_t)(kk + 1) * YD + col];
            cm = wmma_f32(am, bm, cm);
            cl = wmma_f32(al, bl, cl);
        }
        const float bm2 = b2mu[col];
        const float bl2 = b2lv[col];
        for (int v = 0; v < 8; ++v) {
            const int mm = v + 8 * hi;
            mus[mm * YD + col] = cm[v] + bm2;
            lvs[mm * YD + col] = tanhf(cl[v] + bl2);
        }
    }
    __syncthreads();

    // ---- per-row epilogue: each thread handles half a row (64 cols) ----
    const int row = m;
    const int kb  = hi * 64;
    const float* yrow = y + (size_t)(r0 + row) * YD;
    float sCL = 0.0f, sPos = 0.0f;
    for (int k = kb; k < kb + 64; ++k) {
        const float mu = mus[row * YD + k];
        const float lv = lvs[row * YD + k];
        const float iv = expf(-lv);
        const float miv = mu * iv;
        IV[(size_t)(r0 + row) * YD + k] = iv;
        M2[(size_t)(r0 + row) * YD + k] = -2.0f * miv;
        sCL += mu * miv + lv;                     // C_i + L_i contribution
        const float d = mu - yrow[k];
        sPos += -0.5f * d * d * iv - 0.5f * lv;   // positive_i contribution
    }
    atomicAdd(&acc[1], sCL);
    atomicAdd(&acc[2], sPos);
}

// ---------------------------------------------------------------------------
// Kernel 2: dual-GEMM pair sum.  D[i,j] = IV_i . Ysq_j + M2_i . Y_j, reduced
// to a single scalar.  One 16x16 tile per block, 64 fp32 WMMAs (K=128, x2).
// Tiles are contiguous 8 KB blocks -> staged to LDS via async B128 transfers
// (ASYNCcnt path) so fragment feeds become banked LDS reads, not 16-line
// global gathers.
// ---------------------------------------------------------------------------
__global__ __launch_bounds__(32) void pair_gemm_kernel(
    const float* __restrict__ y, const float* __restrict__ IV,
    const float* __restrict__ M2, float* __restrict__ acc) {
    __shared__ float sIV[16 * YD];  // 8 KB: rows it*16 .. it*16+15
    __shared__ float sM2[16 * YD];
    __shared__ float sY[16 * YD];   // rows jt*16 .. jt*16+15

    const int l  = threadIdx.x;
    const int m  = l & 15;
    const int hi = l >> 4;
    const int it = blockIdx.x;
    const int jt = blockIdx.y;

    const float* gIV = IV + (size_t)it * 16 * YD;
    const float* gM2 = M2 + (size_t)it * 16 * YD;
    const float* gY  = y + (size_t)jt * 16 * YD;

    async_tile_16x128(gIV, sIV, l);
    async_tile_16x128(gM2, sM2, l);
    async_tile_16x128(gY, sY, l);
    async_wait0();
    __syncthreads();

    v8f c = {};
    for (int k0 = 0; k0 < YD; k0 += 4) {
        const int kk = k0 + 2 * hi;
        const float y0 = sY[m * YD + kk];      // row j = jt*16 + m (B col), transposed read
        const float y1 = sY[m * YD + kk + 1];
        v2f a1, b1, a2, b2;
        a1.x = sIV[m * YD + kk];               // row i = it*16 + m (A row)
        a1.y = sIV[m * YD + kk + 1];
        b1.x = y0 * y0;                        // Ysq^T fragment
        b1.y = y1 * y1;
        c = wmma_f32(a1, b1, c);
        a2.x = sM2[m * YD + kk];
        a2.y = sM2[m * YD + kk + 1];
        b2.x = y0;                             // Y^T fragment
        b2.y = y1;
        c = wmma_f32(a2, b2, c);
    }
    float s = c[0] + c[1] + c[2] + c[3] + c[4] + c[5] + c[6] + c[7];
    for (int off = 16; off > 0; off >>= 1) s += __shfl_down(s, off, 32);
    if (l == 0) atomicAdd(&acc[0], s);
}

// ---------------------------------------------------------------------------
// Kernel 3: scalar assembly with the exact collapsed logsumexp constant.
// ---------------------------------------------------------------------------
__global__ void finalize_kernel(const float* __restrict__ acc, float* __restrict__ out) {
    const float Bf = (float)BATCH;
    const float S = acc[0], sCL = acc[1], sPos = acc[2];
    const float mean_all = -0.5f * (S / (Bf * Bf) + sCL / Bf);
    // logsumexp over axis 0 collapses: log((B-1) + e^-20) - log(B-1)
    const float coff = logf((Bf - 1.0f) + expf(-20.0f)) - logf(Bf - 1.0f);
    out[0] = sPos / Bf - mean_all - coff;
}

extern "C" void kernel_launch(void* const* d_in, const int* in_sizes, int n_in,
                              void* d_out, int out_size, void* d_ws, size_t ws_size,
                              hipStream_t stream) {
    (void)in_sizes; (void)n_in; (void)out_size; (void)ws_size;
    const float* x    = (const float*)d_in[0];
    const float* y    = (const float*)d_in[1];
    const float* w1mu = (const float*)d_in[2];
    const float* b1mu = (const float*)d_in[3];
    const float* w2mu = (const float*)d_in[4];
    const float* b2mu = (const float*)d_in[5];
    const float* w1lv = (const float*)d_in[6];
    const float* b1lv = (const float*)d_in[7];
    const float* w2lv = (const float*)d_in[8];
    const float* b2lv = (const float*)d_in[9];

    float* ws  = (float*)d_ws;
    float* IV  = ws + IV_OFF;
    float* M2  = ws + M2_OFF;
    float* acc = ws + ACC_OFF;

    (void)hipMemsetAsync(acc, 0, 4 * sizeof(float), stream);  // capture-safe

    mlp_kernel<<<BATCH / 16, 32, 0, stream>>>(x, y, w1mu, b1mu, w2mu, b2mu,
                                              w1lv, b1lv, w2lv, b2lv, IV, M2, acc);
    pair_gemm_kernel<<<dim3(BATCH / 16, BATCH / 16), 32, 0, stream>>>(y, IV, M2, acc);
    finalize_kernel<<<1, 1, 0, stream>>>(acc, (float*)d_out);
}